// LQR_Solver_47571057770743
// MI455X (gfx1250) — compile-verified
//
#include <hip/hip_runtime.h>
#include <hip/hip_bf16.h>

// LQR KKT solve via Riccati recursion on a single persistent workgroup.
// FP32 WMMA (v_wmma_f32_16x16x4_f32) for all tile GEMMs; LDS-resident state;
// C_t double-buffered into LDS via CDNA5 async global->LDS copies (ASYNCcnt).

typedef __attribute__((ext_vector_type(2))) float v2f;
typedef __attribute__((ext_vector_type(8))) float v8f;
typedef int v4i __attribute__((vector_size(16)));   // GCC-style int4 (builtin sig)

#define NS 32            // n_state
#define NI 16            // n_input
#define NA 48            // n_all
#define T_HORIZON 64

#if defined(__has_builtin)
#if __has_builtin(__builtin_amdgcn_global_load_async_to_lds_b128)
#define HAVE_ASYNC_LDS 1
#endif
#if __has_builtin(__builtin_amdgcn_s_wait_asynccnt)
#define HAVE_WAIT_ASYNC 1
#endif
#endif

__device__ __forceinline__ void async_cp16(const float* g, float* l) {
#ifdef HAVE_ASYNC_LDS
    __builtin_amdgcn_global_load_async_to_lds_b128(
        (__attribute__((address_space(1))) v4i*)(g),
        (__attribute__((address_space(3))) v4i*)(l),
        0, 0);
#else
    const float4* gs = (const float4*)g;
    float4* ls = (float4*)l;
    *ls = *gs;
#endif
}

__device__ __forceinline__ void wait_async() {
#ifdef HAVE_ASYNC_LDS
#ifdef HAVE_WAIT_ASYNC
    __builtin_amdgcn_s_wait_asynccnt(0);
#else
    asm volatile("s_wait_asynccnt 0x0" ::: "memory");
#endif
#endif
}

// Copy one 48x48 fp32 cost block (9216 B) global -> LDS as 576 x 16B chunks.
__device__ __forceinline__ void prefetch_C(const float* Ct, float* dst, int tid) {
    for (int idx = tid; idx < (NA * NA) / 4; idx += 256)
        async_cp16(Ct + idx * 4, dst + idx * 4);
}

__global__ __launch_bounds__(256) void lqr_riccati_kernel(
    const float* __restrict__ Ag,   // 32x32
    const float* __restrict__ Bg,   // 32x16
    const float* __restrict__ Cg,   // 64x48x48
    const float* __restrict__ x0g,  // 32
    float* __restrict__ out,        // 5120: tau(3072) then mu(2048)
    float* __restrict__ Kws)        // T*16*32 floats workspace
{
    __shared__ __align__(16) float sC[2][NA * NA]; // double-buffered C_t (18 KB)
    __shared__ float sG[NS * NA];      // [A | B], 32x48
    __shared__ float sP[NS * NS];      // value matrix P, 32x32
    __shared__ float sPG[NS * NA];     // P*G, 32x48
    __shared__ float sQ[NA * NA];      // Q = C_t + G'PG, 48x48
    __shared__ float sAug[NI * NA];    // [Quu | Qux], 16x48
    __shared__ float sKt[NI * NS];     // K_t, 16x32
    __shared__ float sTau[T_HORIZON * NA]; // full tau trajectory (12 KB)
    __shared__ float sVa[NA];
    __shared__ float sVb[NS];
    __shared__ float sF[NI];
    __shared__ float sPiv;

    const int tid  = threadIdx.x;
    const int wave = tid >> 5;         // 8 waves of 32 (wave32)
    const int lane = tid & 31;
    const int half = lane >> 4;        // 0: lanes 0-15, 1: lanes 16-31
    const int lm   = lane & 15;

    // ---- stage in G = [A | B], P = 0; start async prefetch of C_{T-1} ----
    prefetch_C(Cg + (T_HORIZON - 1) * NA * NA, sC[(T_HORIZON - 1) & 1], tid);
    for (int idx = tid; idx < NS * NA; idx += 256) {
        int i = idx / NA, j = idx % NA;
        sG[idx] = (j < NS) ? Ag[i * NS + j] : Bg[i * NI + (j - NS)];
    }
    for (int idx = tid; idx < NS * NS; idx += 256) sP[idx] = 0.0f;
    __syncthreads();

    // =================== backward Riccati sweep ===================
    for (int t = T_HORIZON - 1; t >= 0; --t) {
        // ---- PG = P * G : 2x3 tiles of 16x16, K = 32 ----
        for (int job = wave; job < 6; job += 8) {
            const int r0 = (job / 3) * 16, c0 = (job % 3) * 16;
            v8f acc = {};
            for (int k = 0; k < NS; k += 4) {
                v2f a, b;
                a.x = sP[(r0 + lm) * NS + k + 2 * half];
                a.y = sP[(r0 + lm) * NS + k + 2 * half + 1];
                b.x = sG[(k + 2 * half) * NA + c0 + lm];
                b.y = sG[(k + 2 * half + 1) * NA + c0 + lm];
                acc = __builtin_amdgcn_wmma_f32_16x16x4_f32(
                    false, a, false, b, (short)0, acc, false, false);
            }
            for (int r = 0; r < 8; ++r)
                sPG[(r0 + r + 8 * half) * NA + c0 + lm] = acc[r];
        }
        wait_async();                       // C_t now resident in sC[t&1]
        __syncthreads();

        // kick off async fetch of C_{t-1} into the other buffer
        if (t > 0) prefetch_C(Cg + (t - 1) * NA * NA, sC[(t - 1) & 1], tid);

        // ---- Q = C_t + G' * PG : 3x3 tiles, K = 32 (C_t read from LDS) ----
        const float* Ct = sC[t & 1];
        for (int job = wave; job < 9; job += 8) {
            const int r0 = (job / 3) * 16, c0 = (job % 3) * 16;
            v8f acc;
            for (int r = 0; r < 8; ++r)
                acc[r] = Ct[(r0 + r + 8 * half) * NA + c0 + lm];
            for (int k = 0; k < NS; k += 4) {
                v2f a, b;   // A-op = (G')[r0.., k..] == G[k..][r0..] (transposed read)
                a.x = sG[(k + 2 * half) * NA + r0 + lm];
                a.y = sG[(k + 2 * half + 1) * NA + r0 + lm];
                b.x = sPG[(k + 2 * half) * NA + c0 + lm];
                b.y = sPG[(k + 2 * half + 1) * NA + c0 + lm];
                acc = __builtin_amdgcn_wmma_f32_16x16x4_f32(
                    false, a, false, b, (short)0, acc, false, false);
            }
            for (int r = 0; r < 8; ++r)
                sQ[(r0 + r + 8 * half) * NA + c0 + lm] = acc[r];
        }
        __syncthreads();

        // ---- build augmented [Quu | Qux] (16x48) ----
        for (int idx = tid; idx < NI * NA; idx += 256) {
            int i = idx / NA, j = idx % NA;
            sAug[idx] = (j < NI) ? sQ[(NS + i) * NA + NS + j]
                                 : sQ[(NS + i) * NA + (j - NI)];
        }
        __syncthreads();

        // ---- Gauss-Jordan on SPD Quu (no pivoting) ----
        for (int j = 0; j < NI; ++j) {
            if (tid == 0) sPiv = 1.0f / sAug[j * NA + j];
            __syncthreads();
            if (tid < NA) sAug[j * NA + tid] *= sPiv;
            __syncthreads();
            if (tid < NI) sF[tid] = (tid == j) ? 0.0f : sAug[tid * NA + j];
            __syncthreads();
            for (int idx = tid; idx < NI * NA; idx += 256) {
                int i = idx / NA, c = idx % NA;
                sAug[idx] -= sF[i] * sAug[j * NA + c];
            }
            __syncthreads();
        }

        // ---- K_t = -Quu^-1 Qux ----
        for (int idx = tid; idx < NI * NS; idx += 256) {
            int i = idx / NS, c = idx % NS;
            float kv = -sAug[i * NA + NI + c];
            sKt[idx] = kv;
            Kws[t * (NI * NS) + idx] = kv;
        }
        __syncthreads();

        // ---- P = Qxx + Qxu * K_t : 2x2 tiles, K = 16 ----
        for (int job = wave; job < 4; job += 8) {
            const int r0 = (job >> 1) * 16, c0 = (job & 1) * 16;
            v8f acc;
            for (int r = 0; r < 8; ++r)
                acc[r] = sQ[(r0 + r + 8 * half) * NA + c0 + lm];
            for (int k = 0; k < NI; k += 4) {
                v2f a, b;   // A-op = Qxu = Q[0:32, 32:48]
                a.x = sQ[(r0 + lm) * NA + NS + k + 2 * half];
                a.y = sQ[(r0 + lm) * NA + NS + k + 2 * half + 1];
                b.x = sKt[(k + 2 * half) * NS + c0 + lm];
                b.y = sKt[(k + 2 * half + 1) * NS + c0 + lm];
                acc = __builtin_amdgcn_wmma_f32_16x16x4_f32(
                    false, a, false, b, (short)0, acc, false, false);
            }
            for (int r = 0; r < 8; ++r)
                sP[(r0 + r + 8 * half) * NS + c0 + lm] = acc[r];
        }
        __syncthreads();
    }

    // =================== forward rollout ===================
    if (tid < NS) sVa[tid] = x0g[tid];
    __syncthreads();
    for (int t = 0; t < T_HORIZON; ++t) {
        if (tid < NI) {                       // u_t = K_t x_t
            float s = 0.0f;
            for (int c = 0; c < NS; ++c)
                s += Kws[t * (NI * NS) + tid * NS + c] * sVa[c];
            sVa[NS + tid] = s;
        }
        __syncthreads();
        if (tid < NA) {                       // emit tau_t
            sTau[t * NA + tid] = sVa[tid];
            out[t * NA + tid]  = sVa[tid];
        }
        if (tid < NS) {                       // x_{t+1} = G tau_t
            float s = 0.0f;
            for (int j = 0; j < NA; ++j) s += sG[tid * NA + j] * sVa[j];
            sVb[tid] = s;
        }
        __syncthreads();
        if (tid < NS) sVa[tid] = sVb[tid];
        __syncthreads();
    }

    // =================== costate (mu) backward sweep ===================
    // mu_t = (C_t tau_t)_x + A' mu_{t+1};  mu_{T-1}=(C tau)_x;  mu_0 = -(...)
    prefetch_C(Cg + (T_HORIZON - 1) * NA * NA, sC[(T_HORIZON - 1) & 1], tid);
    for (int t = T_HORIZON - 1; t >= 0; --t) {
        wait_async();
        __syncthreads();
        if (t > 0) prefetch_C(Cg + (t - 1) * NA * NA, sC[(t - 1) & 1], tid);

        float mreg = 0.0f;
        if (tid < NS) {
            const float* Ct = sC[t & 1];
            float g = 0.0f;
            for (int j = 0; j < NA; ++j) g += Ct[tid * NA + j] * sTau[t * NA + j];
            float am = 0.0f;
            if (t < T_HORIZON - 1)
                for (int r = 0; r < NS; ++r) am += Ag[r * NS + tid] * sVb[r];
            mreg = g + am;
            if (t == 0) mreg = -mreg;
            out[T_HORIZON * NA + t * NS + tid] = mreg;
        }
        __syncthreads();
        if (tid < NS) sVb[tid] = mreg;        // becomes mu_{t+1} next iter
    }
}

extern "C" void kernel_launch(void* const* d_in, const int* in_sizes, int n_in,
                              void* d_out, int out_size, void* d_ws, size_t ws_size,
                              hipStream_t stream) {
    const float* A  = (const float*)d_in[0];   // 32x32
    const float* B  = (const float*)d_in[1];   // 32x16
    const float* C  = (const float*)d_in[2];   // 64x48x48
    // d_in[3] = T (scalar, fixed 64)
    const float* x0 = (const float*)d_in[4];   // 32
    float* out = (float*)d_out;
    float* Kws = (float*)d_ws;                 // needs T*16*32*4 = 128 KiB

    lqr_riccati_kernel<<<dim3(1), dim3(256), 0, stream>>>(A, B, C, x0, out, Kws);
}